// DYS_opt_net_37546604102195
// MI455X (gfx1250) — compile-verified
//
#include <hip/hip_runtime.h>

typedef __attribute__((ext_vector_type(2))) float v2f;
typedef __attribute__((ext_vector_type(8))) float v8f;

#define ALPHA 0.05f
#define EPS   0.01f

#define TM 128
#define TN 128
#define TK 32

static __device__ __forceinline__ v8f wmma4(const v2f a, const v2f b, const v8f c)
{
    return __builtin_amdgcn_wmma_f32_16x16x4_f32(false, a, false, b, (short)0, c,
                                                 false, false);
}

static __device__ __forceinline__ void store_tile(
    const v8f acc, int r0, int c0, float* __restrict__ Cm, int ldc,
    const float* __restrict__ Cin, const float* __restrict__ bias,
    float alpha, float beta, float gamma, int relu, int hi, int ln)
{
    const int c = c0 + ln;
    const float bv = bias ? gamma * bias[c] : 0.f;
#pragma unroll
    for (int v = 0; v < 8; ++v) {
        const int r = r0 + v + 8 * hi;
        float val = alpha * acc[v] + bv;
        if (Cin)  val += beta * Cin[(size_t)r * ldc + c];
        if (relu) val = fmaxf(val, 0.f);
        Cm[(size_t)r * ldc + c] = val;
    }
}

// ---------------------------------------------------------------------------
// fp32 WMMA GEMM:  C = alpha*(A @ op(B)) + beta*Cin + gamma*bias[col], opt relu
// op(B) = B (row-major [K,N]) or B^T (B row-major [N,K]) when transB=1.
// Block tile 128x128, K-tile 32. 8 waves in a 2(M) x 4(N) grid; each wave owns
// a 64x32 sub-tile as 4x2 accumulators -> 8 v_wmma_f32_16x16x4_f32 per K-step,
// 64 per K-tile, against ~32 LDS fragment loads (2:1 wmma:ds).
// Requires M%128==0, N%128==0, K%32==0 (all shapes here conform).
// ---------------------------------------------------------------------------
__global__ __launch_bounds__(256) void gemm_f32_wmma(
    const float* __restrict__ Am, int lda,
    const float* __restrict__ Bm, int ldb, int transB,
    float* __restrict__ Cm, int ldc,
    const float* __restrict__ Cin,
    const float* __restrict__ bias,
    float alpha, float beta, float gamma, int relu,
    int K, const int* skip)
{
    if (skip && *skip) return;   // block-uniform: EXEC stays all-ones past here

    __shared__ float As[TM][TK + 4];   // 36-float rows: 16B-aligned, conflict-free
    __shared__ float Bs[TK][TN + 4];   // 132-float rows

    const int t    = threadIdx.x;
    const int row0 = blockIdx.y * TM;
    const int col0 = blockIdx.x * TN;

    const int wid  = t >> 5;
    const int lane = t & 31;
    const int wm   = (wid >> 2) * 64;   // wave row offset: 0 or 64
    const int wn   = (wid & 3) * 32;    // wave col offset: 0,32,64,96
    const int hi   = lane >> 4;
    const int ln   = lane & 15;

    v8f acc[4][2];
#pragma unroll
    for (int mi = 0; mi < 4; ++mi)
#pragma unroll
        for (int ni = 0; ni < 2; ++ni) acc[mi][ni] = (v8f){};

    // cooperative staging indices
    const int arow = t >> 1;            // 0..127
    const int akc  = (t & 1) * 16;      // 0 or 16
    const int bkp  = t >> 3;            // plain-B: 0..31
    const int bnp  = (t & 7) * 16;      // 16 consecutive n
    const int btn  = t >> 1;            // transB: 0..127 (one output column)
    const int btk  = (t & 1) * 16;      // 16 consecutive k

    for (int kb = 0; kb < K; kb += TK) {
        // stage A tile (128x32), four b128 loads/stores per thread
        const float* ap = Am + (size_t)(row0 + arow) * lda + kb + akc;
#pragma unroll
        for (int j = 0; j < 4; ++j)
            *(float4*)&As[arow][akc + 4 * j] = *(const float4*)(ap + 4 * j);
        if (kb + TK < K) __builtin_prefetch(ap + TK, 0, 1);

        if (!transB) {
            const float* bp = Bm + (size_t)(kb + bkp) * ldb + col0 + bnp;
#pragma unroll
            for (int j = 0; j < 4; ++j)
                *(float4*)&Bs[bkp][bnp + 4 * j] = *(const float4*)(bp + 4 * j);
            if (kb + TK < K) __builtin_prefetch(bp + (size_t)TK * ldb, 0, 1);
        } else {
            const float* bp = Bm + (size_t)(col0 + btn) * ldb + kb + btk;
#pragma unroll
            for (int j = 0; j < 4; ++j) {
                const float4 vv = *(const float4*)(bp + 4 * j);
                Bs[btk + 4 * j + 0][btn] = vv.x;
                Bs[btk + 4 * j + 1][btn] = vv.y;
                Bs[btk + 4 * j + 2][btn] = vv.z;
                Bs[btk + 4 * j + 3][btn] = vv.w;
            }
            if (kb + TK < K) __builtin_prefetch(bp + TK, 0, 1);
        }
        __syncthreads();

#pragma unroll
        for (int kk = 0; kk < TK; kk += 4) {
            v2f a[4], b[2];
            // A 16x4 fragment: lanes 0-15 hold K={kk,kk+1}, lanes 16-31 K={kk+2,kk+3}
#pragma unroll
            for (int mi = 0; mi < 4; ++mi) {
                a[mi].x = As[wm + 16 * mi + ln][kk + 2 * hi];
                a[mi].y = As[wm + 16 * mi + ln][kk + 2 * hi + 1];
            }
#pragma unroll
            for (int ni = 0; ni < 2; ++ni) {
                b[ni].x = Bs[kk + 2 * hi][wn + 16 * ni + ln];
                b[ni].y = Bs[kk + 2 * hi + 1][wn + 16 * ni + ln];
            }
#pragma unroll
            for (int mi = 0; mi < 4; ++mi)
#pragma unroll
                for (int ni = 0; ni < 2; ++ni)
                    acc[mi][ni] = wmma4(a[mi], b[ni], acc[mi][ni]);
        }
        __syncthreads();
    }

#pragma unroll
    for (int mi = 0; mi < 4; ++mi)
#pragma unroll
        for (int ni = 0; ni < 2; ++ni)
            store_tile(acc[mi][ni], row0 + wm + 16 * mi, col0 + wn + 16 * ni,
                       Cm, ldc, Cin, bias, alpha, beta, gamma, relu, hi, ln);
}

// ---------------------------------------------------------------------------
// Elementwise / reduction helpers
// ---------------------------------------------------------------------------
__global__ void k_prep_u(const float* __restrict__ z, const float* __restrict__ w,
                         float* __restrict__ u, int n, const int* skip)
{
    if (skip && *skip) return;
    int i = blockIdx.x * blockDim.x + threadIdx.x;
    if (i < n) {
        float zi = z[i];
        float x  = fmaxf(zi, 0.f);
        u[i] = 2.f * x - zi - ALPHA * w[i];   // 2*proj_C1(z) - z - alpha*w
    }
}

__global__ void k_update(float* __restrict__ z, const float* __restrict__ y,
                         float* __restrict__ acc, int n, const int* skip)
{
    __shared__ float sm[256];
    if (*skip) return;                        // block-uniform
    int i = blockIdx.x * blockDim.x + threadIdx.x;
    float d2 = 0.f;
    if (i < n) {
        float zi = z[i];
        float x  = fmaxf(zi, 0.f);
        float yi = y[i];
        z[i] = zi - x + yi;                   // z_{k+1} = z - x + y
        float d = yi - x;                     // z_{k+1} - z
        d2 = d * d;
    }
    sm[threadIdx.x] = d2;
    __syncthreads();
    for (int s = 128; s > 0; s >>= 1) {
        if (threadIdx.x < s) sm[threadIdx.x] += sm[threadIdx.x + s];
        __syncthreads();
    }
    if (threadIdx.x == 0) atomicAdd(acc, sm[0]);
}

__global__ void k_sumsq(const float* __restrict__ x, float* __restrict__ acc, int n)
{
    __shared__ float sm[256];
    int i = blockIdx.x * blockDim.x + threadIdx.x;
    float v = 0.f;
    if (i < n) { float t = x[i]; v = t * t; }
    sm[threadIdx.x] = v;
    __syncthreads();
    for (int s = 128; s > 0; s >>= 1) {
        if (threadIdx.x < s) sm[threadIdx.x] += sm[threadIdx.x + s];
        __syncthreads();
    }
    if (threadIdx.x == 0) atomicAdd(acc, sm[0]);
}

__global__ void k_zero(float* p) { p[0] = 0.f; }

__global__ void k_check(const float* acc, int* flag)
{
    if (*flag == 0 && *acc <= EPS * EPS) *flag = 1;
}

__global__ void k_init_X(float* __restrict__ X, const float* __restrict__ fn, int n)
{
    int i = blockIdx.x * blockDim.x + threadIdx.x;
    if (i < n * n) {
        float c = rsqrtf(*fn);                // X0 = I / ||G||_F  (||G||_F >= lambda_max)
        int r = i / n, cc = i - r * n;
        X[i] = (r == cc) ? c : 0.f;
    }
}

__global__ void k_final(const float* __restrict__ z, const float* __restrict__ y,
                        float* __restrict__ out, int n)
{
    int i = blockIdx.x * blockDim.x + threadIdx.x;
    if (i < n) {
        float zi = z[i];
        float x  = fmaxf(zi, 0.f);
        out[i] = fmaxf(zi - x + y[i], 0.f);   // proj_C1(dys_step(z*, w))
    }
}

// ---------------------------------------------------------------------------
// Host orchestration (graph-capturable: only async ops on `stream`)
// ---------------------------------------------------------------------------
extern "C" void kernel_launch(void* const* d_in, const int* in_sizes, int n_in,
                              void* d_out, int out_size, void* d_ws, size_t ws_size,
                              hipStream_t stream)
{
    const float* ctx  = (const float*)d_in[0];  // d   [256,512]
    const float* Ain  = (const float*)d_in[1];  // A   [1024,4096]
    const float* bvec = (const float*)d_in[2];  // b   [1024]
    const float* W1   = (const float*)d_in[3];  // [512,2048]
    const float* b1   = (const float*)d_in[4];  // [2048]
    const float* W2   = (const float*)d_in[5];  // [2048,4096]
    const float* b2   = (const float*)d_in[6];  // [4096]
    const float* z0   = (const float*)d_in[7];  // [256,4096]
    float* out = (float*)d_out;

    const int B = 256, D = 512, H = 2048, N1 = 1024, N2 = 4096;

    float* W = (float*)d_ws;
    size_t o = 0;
    float* z  = W + o; o += (size_t)B * N2;
    float* u  = W + o; o += (size_t)B * N2;
    float* y  = W + o; o += (size_t)B * N2;
    float* wv = W + o; o += (size_t)B * N2;   // MLP output w
    float* h  = W + o; o += (size_t)B * H;
    float* R  = W + o; o += (size_t)B * N1;   // u@A^T - b
    float* S  = W + o; o += (size_t)B * N1;   // R@Ginv
    float* G  = W + o; o += (size_t)N1 * N1;  // A A^T
    float* Xa = W + o; o += (size_t)N1 * N1;  // Newton-Schulz ping
    float* Xb = W + o; o += (size_t)N1 * N1;  // Newton-Schulz pong
    float* T1 = W + o; o += (size_t)N1 * N1;  // G @ X
    float* acc = W + o; o += 1;
    float* fn  = W + o; o += 1;
    int*   flag = (int*)(W + o); o += 1;

    dim3 blk(256);
    auto gemm = [&](const float* A_, int lda, const float* B_, int ldb, int tB,
                    float* C_, int ldc, const float* Cin, const float* bias,
                    float al, float be, float ga, int relu,
                    int M, int N, int K, const int* skip) {
        dim3 grid(N / TN, M / TM);
        gemm_f32_wmma<<<grid, blk, 0, stream>>>(A_, lda, B_, ldb, tB, C_, ldc,
                                                Cin, bias, al, be, ga, relu, K, skip);
    };

    // init: z = z0, flag = 0
    hipMemcpyAsync(z, z0, sizeof(float) * (size_t)B * N2, hipMemcpyDeviceToDevice, stream);
    hipMemsetAsync(flag, 0, sizeof(int), stream);

    // w = relu(d@W1 + b1)@W2 + b2
    gemm(ctx, D, W1, H, 0, h,  H,  nullptr, b1, 1.f, 0.f, 1.f, 1, B, H,  D, nullptr);
    gemm(h,   H, W2, N2, 0, wv, N2, nullptr, b2, 1.f, 0.f, 1.f, 0, B, N2, H, nullptr);

    // G = A @ A^T  (pseudo-inverse path: A+ = A^T G^-1)
    gemm(Ain, N2, Ain, N2, 1, G, N1, nullptr, nullptr, 1.f, 0.f, 0.f, 0, N1, N1, N2, nullptr);

    // Ginv via Newton-Schulz: X0 = I/||G||_F, X <- 2X - X(GX), 16 iters
    k_zero<<<1, 1, 0, stream>>>(fn);
    k_sumsq<<<dim3((N1 * N1 + 255) / 256), blk, 0, stream>>>(G, fn, N1 * N1);
    k_init_X<<<dim3((N1 * N1 + 255) / 256), blk, 0, stream>>>(Xa, fn, N1);

    float* Xc = Xa; float* Xn = Xb;
    for (int it = 0; it < 16; ++it) {
        gemm(G,  N1, Xc, N1, 0, T1, N1, nullptr, nullptr,  1.f, 0.f, 0.f, 0, N1, N1, N1, nullptr);
        gemm(Xc, N1, T1, N1, 0, Xn, N1, Xc,      nullptr, -1.f, 2.f, 0.f, 0, N1, N1, N1, nullptr);
        float* tmp = Xc; Xc = Xn; Xn = tmp;
    }
    float* Ginv = Xc;

    // fixed-point loop: 50 gated iterations emulate while(diff>EPS && k<50)
    const int nz = B * N2;
    dim3 gout((nz + 255) / 256);
    for (int itr = 0; itr < 50; ++itr) {
        k_prep_u<<<gout, blk, 0, stream>>>(z, wv, u, nz, (const int*)flag);
        gemm(u, N2, Ain,  N2, 1, R, N1, nullptr, bvec, 1.f, 0.f, -1.f, 0, B, N1, N2, flag);
        gemm(R, N1, Ginv, N1, 0, S, N1, nullptr, nullptr, 1.f, 0.f, 0.f, 0, B, N1, N1, flag);
        gemm(S, N1, Ain,  N2, 0, y, N2, u, nullptr, -1.f, 1.f, 0.f, 0, B, N2, N1, flag);
        k_zero<<<1, 1, 0, stream>>>(acc);
        k_update<<<gout, blk, 0, stream>>>(z, y, acc, nz, (const int*)flag);
        k_check<<<1, 1, 0, stream>>>((const float*)acc, flag);
    }

    // final differentiable step on z*, then project_C1
    k_prep_u<<<gout, blk, 0, stream>>>(z, wv, u, nz, (const int*)nullptr);
    gemm(u, N2, Ain,  N2, 1, R, N1, nullptr, bvec, 1.f, 0.f, -1.f, 0, B, N1, N2, nullptr);
    gemm(R, N1, Ginv, N1, 0, S, N1, nullptr, nullptr, 1.f, 0.f, 0.f, 0, B, N1, N1, nullptr);
    gemm(S, N1, Ain,  N2, 0, y, N2, u, nullptr, -1.f, 1.f, 0.f, 0, B, N2, N1, nullptr);
    k_final<<<gout, blk, 0, stream>>>(z, y, out, nz);
}